// NoAttentionNet_9990093930998
// MI455X (gfx1250) — compile-verified
//
#include <hip/hip_runtime.h>
#include <hip/hip_bf16.h>

typedef __attribute__((ext_vector_type(2))) float v2f;
typedef __attribute__((ext_vector_type(8))) float v8f;

#define N_NODES 50000
#define N_EDGES 800000
#define N_GRAPHS 512
#define IN_CH 128
#define HID 64

__device__ __forceinline__ float selu_f(float x) {
    const float scale = 1.0507009873554804934193349852946f;
    const float alpha = 1.6732632423543772848170429916717f;
    return scale * (x > 0.0f ? x : alpha * (expf(x) - 1.0f));
}

// ---------------- zero fill ----------------
__global__ void zero_f32_k(float* __restrict__ p, int n) {
    int i = blockIdx.x * blockDim.x + threadIdx.x;
    if (i < n) p[i] = 0.0f;
}

// ---------------- WMMA f32 GEMM: Y[rows x 64] = X[rows x in_ch] @ W[in_ch x 64] ----------------
// One wave computes one 16x16 tile of Y using V_WMMA_F32_16X16X4_F32.
// rows must be a multiple of 16 (50000 = 3125*16).
__global__ void gemm_wmma_f32(const float* __restrict__ X,
                              const float* __restrict__ W,
                              float* __restrict__ Y,
                              int rows, int in_ch) {
    const int lane = threadIdx.x & 31;
    const int wave = (blockIdx.x * (blockDim.x >> 5)) + (threadIdx.x >> 5);
    const int tiles_m = rows >> 4;          // 16-row tiles
    const int tm = wave >> 2;               // 4 col-tiles (64/16)
    const int tn = wave & 3;
    if (tm >= tiles_m) return;              // wave-uniform: EXEC stays all-1s

    const int m0 = tm << 4;
    const int n0 = tn << 4;
    const int half = lane >> 4;             // 0: lanes 0-15, 1: lanes 16-31
    const int l15 = lane & 15;

    const float* xrow = X + (size_t)(m0 + l15) * in_ch;
    const int col = n0 + l15;

    v8f c = {0.f, 0.f, 0.f, 0.f, 0.f, 0.f, 0.f, 0.f};

    for (int k = 0; k < in_ch; k += 4) {
        const int ka = k + half * 2;        // lanes 0-15: K=k,k+1 ; lanes 16-31: K=k+2,k+3
        // A 16x4 fragment: VGPR0 = K even, VGPR1 = K odd (per ISA layout)
        v2f a = *(const v2f*)(xrow + ka);
        // B 4x16 fragment: row K striped across lanes (mirrored layout)
        v2f b;
        b.x = W[(size_t)ka * HID + col];
        b.y = W[(size_t)(ka + 1) * HID + col];
        c = __builtin_amdgcn_wmma_f32_16x16x4_f32(false, a, false, b,
                                                  (short)0, c, false, false);
    }

    // C/D layout: VGPR r -> row m0 + r + 8*half, col = n0 + (lane&15)
    const int rbase = m0 + half * 8;
#pragma unroll
    for (int r = 0; r < 8; ++r) {
        Y[(size_t)(rbase + r) * HID + col] = c[r];
    }
}

// ---------------- edge scatter: agg[dst] += y[src], 64 channels ----------------
// 16 threads per edge, 4 channels each (float4 load + 4 f32 atomics).
__global__ void scatter_edges_k(const float* __restrict__ y,
                                const int* __restrict__ src,
                                const int* __restrict__ dst,
                                float* __restrict__ agg, int n_edges) {
    long long tid = (long long)blockIdx.x * blockDim.x + threadIdx.x;
    int e = (int)(tid >> 4);
    if (e >= n_edges) return;
    int c4 = ((int)tid & 15) << 2;
    int s = src[e];
    int d = dst[e];
    const float4 v = *(const float4*)(y + (size_t)s * HID + c4);
    float* p = agg + (size_t)d * HID + c4;
    atomicAdd(p + 0, v.x);
    atomicAdd(p + 1, v.y);
    atomicAdd(p + 2, v.z);
    atomicAdd(p + 3, v.w);
}

// ---------------- elementwise: h += bias (broadcast over 64 ch) ----------------
__global__ void add_bias_k(float* __restrict__ h, const float* __restrict__ b, int n) {
    int i = blockIdx.x * blockDim.x + threadIdx.x;
    if (i < n) h[i] += b[i & (HID - 1)];
}

// ---------------- elementwise: h = selu(h + bias) ----------------
__global__ void selu_bias_k(float* __restrict__ h, const float* __restrict__ b, int n) {
    int i = blockIdx.x * blockDim.x + threadIdx.x;
    if (i < n) h[i] = selu_f(h[i] + b[i & (HID - 1)]);
}

// ---------------- gate = h2 @ gate_W + gate_b ; atomicMax per graph ----------------
__device__ __forceinline__ unsigned map_f2u(float f) {
    unsigned u = __float_as_uint(f);
    return (u >> 31) ? ~u : (u | 0x80000000u);
}
__device__ __forceinline__ float unmap_u2f(unsigned u) {
    return (u >> 31) ? __uint_as_float(u & 0x7FFFFFFFu) : __uint_as_float(~u);
}

__global__ void gate_k(const float* __restrict__ h2,
                       const float* __restrict__ gateW,
                       const float* __restrict__ gateb,
                       const int* __restrict__ gid,
                       float* __restrict__ gate,
                       unsigned* __restrict__ gmaxu, int n_nodes) {
    int n = blockIdx.x * blockDim.x + threadIdx.x;
    if (n >= n_nodes) return;
    const float* hr = h2 + (size_t)n * HID;
    float acc = gateb[0];
#pragma unroll
    for (int c = 0; c < HID; ++c) acc += hr[c] * gateW[c];
    gate[n] = acc;
    atomicMax(gmaxu + gid[n], map_f2u(acc));
}

// ---------------- e = exp(gate - gmax[g]) ; denom[g] += e ----------------
__global__ void exp_k(const float* __restrict__ gate,
                      const unsigned* __restrict__ gmaxu,
                      const int* __restrict__ gid,
                      float* __restrict__ ebuf,
                      float* __restrict__ denom, int n_nodes) {
    int n = blockIdx.x * blockDim.x + threadIdx.x;
    if (n >= n_nodes) return;
    int g = gid[n];
    float ev = expf(gate[n] - unmap_u2f(gmaxu[g]));
    ebuf[n] = ev;
    atomicAdd(denom + g, ev);
}

// ---------------- readout[g] += (e/denom) * h2[n], 16 threads/node ----------------
__global__ void readout_k(const float* __restrict__ h2,
                          const float* __restrict__ ebuf,
                          const float* __restrict__ denom,
                          const int* __restrict__ gid,
                          float* __restrict__ readout, int n_nodes) {
    long long tid = (long long)blockIdx.x * blockDim.x + threadIdx.x;
    int n = (int)(tid >> 4);
    if (n >= n_nodes) return;
    int c4 = ((int)tid & 15) << 2;
    int g = gid[n];
    float a = ebuf[n] / denom[g];
    const float4 v = *(const float4*)(h2 + (size_t)n * HID + c4);
    float* p = readout + (size_t)g * HID + c4;
    atomicAdd(p + 0, a * v.x);
    atomicAdd(p + 1, a * v.y);
    atomicAdd(p + 2, a * v.z);
    atomicAdd(p + 3, a * v.w);
}

// ---------------- final MLP: one thread per graph ----------------
__global__ void mlp_k(const float* __restrict__ readout,
                      const float* __restrict__ fc1W, const float* __restrict__ fc1b,
                      const float* __restrict__ fc2W, const float* __restrict__ fc2b,
                      float* __restrict__ out, int n_graphs) {
    int g = blockIdx.x * blockDim.x + threadIdx.x;
    if (g >= n_graphs) return;
    const float* r = readout + (size_t)g * HID;
    float z[HID / 2];
#pragma unroll 4
    for (int j = 0; j < HID / 2; ++j) {
        float acc = fc1b[j];
        for (int i = 0; i < HID; ++i) acc += r[i] * fc1W[i * (HID / 2) + j];
        z[j] = selu_f(acc);
    }
    float acc = fc2b[0];
#pragma unroll
    for (int j = 0; j < HID / 2; ++j) acc += z[j] * fc2W[j];
    out[g] = 1.0f / (1.0f + expf(-acc));
}

extern "C" void kernel_launch(void* const* d_in, const int* in_sizes, int n_in,
                              void* d_out, int out_size, void* d_ws, size_t ws_size,
                              hipStream_t stream) {
    const float* x      = (const float*)d_in[0];   // [N_NODES, 128]
    const int*   src    = (const int*)d_in[1];     // [N_EDGES]
    const int*   dst    = (const int*)d_in[2];     // [N_EDGES]
    const int*   gid    = (const int*)d_in[3];     // [N_NODES]
    const float* W1     = (const float*)d_in[4];   // [128, 64]
    const float* b1     = (const float*)d_in[5];   // [64]
    const float* W2     = (const float*)d_in[6];   // [64, 64]
    const float* b2     = (const float*)d_in[7];   // [64]
    const float* gateW  = (const float*)d_in[8];   // [64, 1]
    const float* gateb  = (const float*)d_in[9];   // [1]
    const float* fc1W   = (const float*)d_in[10];  // [64, 32]
    const float* fc1b   = (const float*)d_in[11];  // [32]
    const float* fc2W   = (const float*)d_in[12];  // [32, 1]
    const float* fc2b   = (const float*)d_in[13];  // [1]
    float* out = (float*)d_out;                    // [512, 1]

    // Workspace layout (floats)
    const size_t NB = (size_t)N_NODES * HID;       // 3,200,000
    float* bufA   = (float*)d_ws;                  // y1, later agg2/h2
    float* bufB   = bufA + NB;                     // agg1/h1
    float* bufC   = bufB + NB;                     // y2
    float* gate   = bufC + NB;                     // [N_NODES]
    float* ebuf   = gate + N_NODES;                // [N_NODES]
    float* smalls = ebuf + N_NODES;                // gmaxu(512) + denom(512) + readout(512*64)
    unsigned* gmaxu = (unsigned*)smalls;
    float* denom  = smalls + N_GRAPHS;
    float* readout = denom + N_GRAPHS;
    const int n_small = N_GRAPHS + N_GRAPHS + N_GRAPHS * HID;

    const int nelem = (int)NB;
    dim3 blk(256);

    // 1) zero agg1 + pooling accumulators
    zero_f32_k<<<(nelem + 255) / 256, blk, 0, stream>>>(bufB, nelem);
    zero_f32_k<<<(n_small + 255) / 256, blk, 0, stream>>>(smalls, n_small);

    // 2) y1 = x @ W1   (transform-before-aggregate: valid since segment_sum is linear)
    {
        int tiles = (N_NODES / 16) * (HID / 16);     // 12500 waves
        int blocks = (tiles + 7) / 8;                // 8 waves / 256-thread block
        gemm_wmma_f32<<<blocks, blk, 0, stream>>>(x, W1, bufA, N_NODES, IN_CH);
    }

    // 3) agg1[dst] += y1[src]
    {
        long long th = (long long)N_EDGES * 16;
        scatter_edges_k<<<(int)((th + 255) / 256), blk, 0, stream>>>(bufA, src, dst, bufB, N_EDGES);
    }

    // 4) h1 = agg1 + b1
    add_bias_k<<<(nelem + 255) / 256, blk, 0, stream>>>(bufB, b1, nelem);

    // 5) y2 = h1 @ W2
    {
        int tiles = (N_NODES / 16) * (HID / 16);
        int blocks = (tiles + 7) / 8;
        gemm_wmma_f32<<<blocks, blk, 0, stream>>>(bufB, W2, bufC, N_NODES, HID);
    }

    // 6) zero agg2 (reuse bufA)
    zero_f32_k<<<(nelem + 255) / 256, blk, 0, stream>>>(bufA, nelem);

    // 7) agg2[dst] += y2[src]
    {
        long long th = (long long)N_EDGES * 16;
        scatter_edges_k<<<(int)((th + 255) / 256), blk, 0, stream>>>(bufC, src, dst, bufA, N_EDGES);
    }

    // 8) h2 = selu(agg2 + b2)  (in place in bufA)
    selu_bias_k<<<(nelem + 255) / 256, blk, 0, stream>>>(bufA, b2, nelem);

    // 9) gate + per-graph max
    gate_k<<<(N_NODES + 255) / 256, blk, 0, stream>>>(bufA, gateW, gateb, gid, gate, gmaxu, N_NODES);

    // 10) exp + per-graph denom
    exp_k<<<(N_NODES + 255) / 256, blk, 0, stream>>>(gate, gmaxu, gid, ebuf, denom, N_NODES);

    // 11) attention-weighted readout
    {
        long long th = (long long)N_NODES * 16;
        readout_k<<<(int)((th + 255) / 256), blk, 0, stream>>>(bufA, ebuf, denom, gid, readout, N_NODES);
    }

    // 12) final MLP -> sigmoid
    mlp_k<<<(N_GRAPHS + 255) / 256, blk, 0, stream>>>(readout, fc1W, fc1b, fc2W, fc2b, out, N_GRAPHS);
}